// FusedSqueezeExcitation_79688823210440
// MI455X (gfx1250) — compile-verified
//
#include <hip/hip_runtime.h>

typedef __attribute__((ext_vector_type(2))) float v2f;
typedef __attribute__((ext_vector_type(4))) float v4f;
typedef __attribute__((ext_vector_type(8))) float v8f;

#define SE_N   64
#define SE_C   960
#define SE_CS  240
#define SE_HW  784      // 28*28
#define SE_HW4 196      // 784/4

// ---------------------------------------------------------------------------
// Kernel 1: global average pool.  One wave32 per (n,c) row of 784 floats.
// x reads use default (RT) temporal hint so the 193MB tensor lands in the
// 192MB L2, ready for the re-read in se_scale.
// ---------------------------------------------------------------------------
__global__ __launch_bounds__(256) void se_pool(const float* __restrict__ x,
                                               float* __restrict__ pooled) {
    unsigned wave = (blockIdx.x * blockDim.x + threadIdx.x) >> 5;
    unsigned lane = threadIdx.x & 31u;
    if (wave >= SE_N * SE_C) return;
    const v4f* row = (const v4f*)(x + (size_t)wave * SE_HW);
    float s = 0.0f;
    for (unsigned i = lane; i < SE_HW4; i += 32u) {
        v4f v = row[i];
        s += v.x + v.y + v.z + v.w;
    }
    // wave32 butterfly reduction
    #pragma unroll
    for (int off = 16; off > 0; off >>= 1)
        s += __shfl_xor(s, off, 32);
    if (lane == 0) pooled[wave] = s * (1.0f / SE_HW);
}

// ---------------------------------------------------------------------------
// WMMA fragment helpers for V_WMMA_F32_16X16X4_F32 (wave32).
// A (16x4 f32): lanes 0-15 hold M=lane, lanes 16-31 hold M=lane-16.
//   VGPR0 = K = (lane<16 ? 0 : 2), VGPR1 = K = (lane<16 ? 1 : 3).
// B (4x16 f32): mirrored — lane%16 indexes N, lane half selects K pair.
// C/D (16x16 f32, 8 VGPRs): VGPR r, lanes 0-15 -> M=r, N=lane;
//   lanes 16-31 -> M=r+8, N=lane-16.
// ---------------------------------------------------------------------------

// Kernel 2: h = relu(pooled @ w1^T + b1).   M=64, N=240 (Cs), K=960 (C).
// One wave per 16x16 output tile; grid = (64/16)*(240/16) = 60 blocks of 32.
__global__ __launch_bounds__(32) void se_fc1(const float* __restrict__ pooled,
                                             const float* __restrict__ w1,
                                             const float* __restrict__ b1,
                                             float* __restrict__ h) {
    const int NT = SE_CS / 16;            // 15
    int mt = blockIdx.x / NT;
    int nt = blockIdx.x % NT;
    int lane = threadIdx.x;
    int half = lane >> 4;                 // 0: K{0,1}, 1: K{2,3}
    int l16  = lane & 15;
    int m0 = mt * 16, n0 = nt * 16;

    v8f acc = {};
    const float* arow = pooled + (size_t)(m0 + l16) * SE_C + half * 2;
    const float* brow = w1     + (size_t)(n0 + l16) * SE_C + half * 2;  // B[k][n] = w1[n][k]
    for (int k0 = 0; k0 < SE_C; k0 += 4) {
        v2f a, b;
        a.x = arow[k0];  a.y = arow[k0 + 1];
        b.x = brow[k0];  b.y = brow[k0 + 1];
        acc = __builtin_amdgcn_wmma_f32_16x16x4_f32(
            /*neg_a=*/false, a, /*neg_b=*/false, b,
            /*c_mod=*/(short)0, acc, /*reuse_a=*/false, /*reuse_b=*/false);
    }

    int n = n0 + l16;
    float bias = b1[n];
    #pragma unroll
    for (int r = 0; r < 8; ++r) {
        int m = m0 + r + half * 8;
        float v = acc[r] + bias;
        h[(size_t)m * SE_CS + n] = v > 0.0f ? v : 0.0f;
    }
}

// Kernel 3: scale = hardsigmoid(h @ w2^T + b2).  M=64, N=960 (C), K=240 (Cs).
// Grid = (64/16)*(960/16) = 240 blocks of 32.
__global__ __launch_bounds__(32) void se_fc2(const float* __restrict__ h,
                                             const float* __restrict__ w2,
                                             const float* __restrict__ b2,
                                             float* __restrict__ scale) {
    const int NT = SE_C / 16;             // 60
    int mt = blockIdx.x / NT;
    int nt = blockIdx.x % NT;
    int lane = threadIdx.x;
    int half = lane >> 4;
    int l16  = lane & 15;
    int m0 = mt * 16, n0 = nt * 16;

    v8f acc = {};
    const float* arow = h  + (size_t)(m0 + l16) * SE_CS + half * 2;
    const float* brow = w2 + (size_t)(n0 + l16) * SE_CS + half * 2;  // B[k][n] = w2[n][k]
    for (int k0 = 0; k0 < SE_CS; k0 += 4) {
        v2f a, b;
        a.x = arow[k0];  a.y = arow[k0 + 1];
        b.x = brow[k0];  b.y = brow[k0 + 1];
        acc = __builtin_amdgcn_wmma_f32_16x16x4_f32(
            false, a, false, b, (short)0, acc, false, false);
    }

    int n = n0 + l16;
    float bias = b2[n];
    #pragma unroll
    for (int r = 0; r < 8; ++r) {
        int m = m0 + r + half * 8;
        float v = acc[r] + bias;
        float g = (v + 3.0f) * (1.0f / 6.0f);
        g = fminf(fmaxf(g, 0.0f), 1.0f);
        scale[(size_t)m * SE_C + n] = g;
    }
}

// ---------------------------------------------------------------------------
// Kernel 4: out[n,c,h,w] = scale[n,c] * x[n,c,h,w].  float4 path; the x
// re-read should hit L2 (primed by se_pool).  Nontemporal stores keep the
// output from evicting x.
// ---------------------------------------------------------------------------
__global__ __launch_bounds__(256) void se_scale(const float* __restrict__ x,
                                                const float* __restrict__ scale,
                                                float* __restrict__ out) {
    unsigned tid = blockIdx.x * blockDim.x + threadIdx.x;
    const unsigned total4 = SE_N * SE_C * SE_HW4;   // 12,042,240
    if (tid >= total4) return;
    unsigned row = tid / SE_HW4;                    // (n*C + c)
    float s = scale[row];
    v4f v = ((const v4f*)x)[tid];
    v.x *= s; v.y *= s; v.z *= s; v.w *= s;
    __builtin_nontemporal_store(v, ((v4f*)out) + tid);
}

extern "C" void kernel_launch(void* const* d_in, const int* in_sizes, int n_in,
                              void* d_out, int out_size, void* d_ws, size_t ws_size,
                              hipStream_t stream) {
    const float* x  = (const float*)d_in[0];
    const float* w1 = (const float*)d_in[1];
    const float* b1 = (const float*)d_in[2];
    const float* w2 = (const float*)d_in[3];
    const float* b2 = (const float*)d_in[4];
    float* out = (float*)d_out;

    // scratch layout in d_ws
    float* pooled = (float*)d_ws;                 // 64*960  = 61440 floats
    float* h      = pooled + SE_N * SE_C;         // 64*240  = 15360 floats
    float* scl    = h + SE_N * SE_CS;             // 64*960  = 61440 floats

    // K1: pool — one wave per row, 8 waves per block
    {
        int rows = SE_N * SE_C;                   // 61440
        int blocks = rows / 8;                    // 7680
        se_pool<<<blocks, 256, 0, stream>>>(x, pooled);
    }
    // K2: fc1 + relu — 4*15 = 60 tiles, one wave each
    se_fc1<<<(SE_N / 16) * (SE_CS / 16), 32, 0, stream>>>(pooled, w1, b1, h);
    // K3: fc2 + hardsigmoid — 4*60 = 240 tiles
    se_fc2<<<(SE_N / 16) * (SE_C / 16), 32, 0, stream>>>(h, w2, b2, scl);
    // K4: broadcast multiply
    {
        unsigned total4 = SE_N * SE_C * SE_HW4;   // 12,042,240
        unsigned blocks = (total4 + 255u) / 256u; // 47040
        se_scale<<<blocks, 256, 0, stream>>>(x, scl, out);
    }
}